// ParabolicIntegrate_53953379172772
// MI455X (gfx1250) — compile-verified
//
#include <hip/hip_runtime.h>

// ---------------------------------------------------------------------------
// ParabolicIntegrate on gfx1250 (MI455X):
//   out[b,t,n,{dW,f1,f2,f3,f4}], causal recurrences c_t = (c_{t-1}+x_t)@M.
// Design (latency-bound serial recurrence, all state on-chip):
//   * batch rows independent across the recurrence -> 4 WGs x 16 rows, no
//     inter-WG traffic
//   * M column-block (128x16 f32) resident in VGPRs per wave (64 VGPRs)
//   * states C1..C4 in LDS (pitch 132 -> conflict-free ds_load_b64)
//   * dW double-buffered in LDS; step t+1's slice is staged during step t's
//     GEMM1 so the global-load latency hides under the WMMA chain
//   * GEMM1 uses two accumulators to break the 32-deep dependent WMMA chain
//   * exact-precision V_WMMA_F32_16X16X4_F32 (bf16 would drift over 63 steps)
// ---------------------------------------------------------------------------

typedef __attribute__((ext_vector_type(2))) float v2f;
typedef __attribute__((ext_vector_type(8))) float v8f;

struct __attribute__((packed)) f3 { float x, y, z; };  // 12-byte store

namespace {
constexpr int kN     = 128;  // spatial points
constexpr int kT     = 64;   // time points
constexpr int kRows  = 16;   // batch rows per workgroup
constexpr int kPitch = 132;  // padded LDS row stride (floats)
}

__global__ __launch_bounds__(256, 1)
void parabolic_integrate_wmma(const float* __restrict__ W,
                              const float* __restrict__ M,
                              float* __restrict__ out) {
  __shared__ float C1[kRows * kPitch];
  __shared__ float C2[kRows * kPitch];
  __shared__ float C3[kRows * kPitch];
  __shared__ float C4[kRows * kPitch];
  __shared__ float DW[2][kRows * kPitch];   // double-buffered dW slice

  const int tid  = threadIdx.x;
  const int lane = tid & 31;
  const int wv   = tid >> 5;       // wave id 0..7 -> output n-tile
  const int half = lane >> 4;      // half-wave select (K split / row split)
  const int lr   = lane & 15;      // lane-in-half: A row / B,C,D column
  const int col  = wv * 16 + lr;   // output column 0..127
  const int b0   = blockIdx.x * kRows;

  // staging assignment: thread handles 8 elements e = tid + i*256 of a slice
  int srow[8], scol[8];
#pragma unroll
  for (int i = 0; i < 8; ++i) {
    const int e = tid + i * 256;
    srow[i] = e >> 7;
    scol[i] = e & 127;
  }

  // ---- B operand: column block of M, resident in VGPRs for the whole run.
  // V_WMMA_F32_16X16X4_F32 lane layout: K = 4*kb + 2*half + {0,1}.
  v2f Breg[32];
#pragma unroll
  for (int kb = 0; kb < 32; ++kb) {
    const int k0 = kb * 4 + half * 2;
    Breg[kb].x = M[(k0 + 0) * kN + col];
    Breg[kb].y = M[(k0 + 1) * kN + col];
  }

  // ---- zero state (c_0 = 0) and the t=0 output block (all channels 0)
  for (int e = tid; e < kRows * kPitch; e += 256) {
    C1[e] = 0.0f; C2[e] = 0.0f; C3[e] = 0.0f; C4[e] = 0.0f;
  }
  for (int e = tid; e < kRows * kN * 5; e += 256) {
    const int r   = e / (kN * 5);
    const int rem = e % (kN * 5);
    out[((size_t)(b0 + r) * kT + 0) * (kN * 5) + rem] = 0.0f;
  }

  // ---- prime the pipeline: wprev = W[:,1,:]; DW[0] = dW_1; emit ch0 @ t=1
  float wprev[8];
#pragma unroll
  for (int i = 0; i < 8; ++i) {
    const size_t g0 = ((size_t)(b0 + srow[i]) * kT + 0) * kN + scol[i];
    const float w0 = W[g0];
    const float w1 = W[g0 + kN];
    const float d  = w1 - w0;
    wprev[i] = w1;
    DW[0][srow[i] * kPitch + scol[i]] = d;
    out[(g0 + kN) * 5 + 0] = d;
  }
  __syncthreads();

  for (int t = 1; t < kT; ++t) {
    const int cb = (t - 1) & 1;   // dW buffer for this step
    const int nb = t & 1;         // dW buffer being staged for step t+1

    // -- Phase A: issue next W-slice loads, run GEMM1, then stage dW_{t+1}.
    float wnext[8];
    const bool stage = (t + 1 < kT);
    if (stage) {
#pragma unroll
      for (int i = 0; i < 8; ++i)
        wnext[i] = W[((size_t)(b0 + srow[i]) * kT + (t + 1)) * kN + scol[i]];
    }

    // GEMM1: Y1 = (C1 + dW_t) @ M — two accumulators break the serial chain
    v8f acc1 = {}, acc1b = {};
#pragma unroll
    for (int kb = 0; kb < 32; kb += 2) {
      const int k0 = kb * 4 + half * 2;
      const v2f a0 = *(const v2f*)&C1[lr * kPitch + k0] +
                     *(const v2f*)&DW[cb][lr * kPitch + k0];
      const v2f a1 = *(const v2f*)&C1[lr * kPitch + k0 + 4] +
                     *(const v2f*)&DW[cb][lr * kPitch + k0 + 4];
      acc1  = __builtin_amdgcn_wmma_f32_16x16x4_f32(
          false, a0, false, Breg[kb],     (short)0, acc1,  false, false);
      acc1b = __builtin_amdgcn_wmma_f32_16x16x4_f32(
          false, a1, false, Breg[kb + 1], (short)0, acc1b, false, false);
    }
#pragma unroll
    for (int v = 0; v < 8; ++v) acc1[v] += acc1b[v];

    if (stage) {
#pragma unroll
      for (int i = 0; i < 8; ++i) {
        const float d = wnext[i] - wprev[i];
        wprev[i] = wnext[i];
        DW[nb][srow[i] * kPitch + scol[i]] = d;
        out[(((size_t)(b0 + srow[i]) * kT + (t + 1)) * kN + scol[i]) * 5 + 0] = d;
      }
    }
    __syncthreads();   // C1 reads done; DW[nb] published for next step

    // -- Phase B: commit Y1 -> C1, emit channel 1
#pragma unroll
    for (int v = 0; v < 8; ++v) {
      const int row = v + 8 * half;
      C1[row * kPitch + col] = acc1[v];
      out[(((size_t)(b0 + row) * kT + t) * kN + col) * 5 + 1] = acc1[v];
    }
    __syncthreads();

    // -- Phase C: GEMMs 2/3/4, integrands from the fresh Y1 tile (3-way ILP)
    v8f acc2 = {}, acc3 = {}, acc4 = {};
#pragma unroll
    for (int kb = 0; kb < 32; ++kb) {
      const int k0 = kb * 4 + half * 2;
      const v2f y  = *(const v2f*)&C1[lr * kPitch + k0];
      const v2f d  = *(const v2f*)&DW[cb][lr * kPitch + k0];
      const v2f y2 = y * y;
      const v2f a2 = *(const v2f*)&C2[lr * kPitch + k0] + y2;
      const v2f a3 = *(const v2f*)&C3[lr * kPitch + k0] + y2 * y;
      const v2f a4 = *(const v2f*)&C4[lr * kPitch + k0] + d * y2;
      const v2f b  = Breg[kb];
      acc2 = __builtin_amdgcn_wmma_f32_16x16x4_f32(
          false, a2, false, b, (short)0, acc2, false, false);
      acc3 = __builtin_amdgcn_wmma_f32_16x16x4_f32(
          false, a3, false, b, (short)0, acc3, false, false);
      acc4 = __builtin_amdgcn_wmma_f32_16x16x4_f32(
          false, a4, false, b, (short)0, acc4, false, false);
    }
    __syncthreads();   // C2..C4 / DW[cb] reads done before overwrite

    // -- Phase D: commit states, emit channels 2/3/4 as one 12-byte store
#pragma unroll
    for (int v = 0; v < 8; ++v) {
      const int row = v + 8 * half;
      const size_t gi = (((size_t)(b0 + row) * kT + t) * kN + col) * 5;
      C2[row * kPitch + col] = acc2[v];
      C3[row * kPitch + col] = acc3[v];
      C4[row * kPitch + col] = acc4[v];
      f3 pkt = { acc2[v], acc3[v], acc4[v] };
      *(f3*)&out[gi + 2] = pkt;
    }
    __syncthreads();
  }
}

extern "C" void kernel_launch(void* const* d_in, const int* in_sizes, int n_in,
                              void* d_out, int out_size, void* d_ws, size_t ws_size,
                              hipStream_t stream) {
  (void)in_sizes; (void)n_in; (void)out_size; (void)d_ws; (void)ws_size;
  const float* W = (const float*)d_in[0];   // [64, 64, 128] f32
  const float* M = (const float*)d_in[1];   // [128, 128]    f32
  float* out = (float*)d_out;               // [64, 64, 128, 5] f32
  parabolic_integrate_wmma<<<dim3(64 / kRows), dim3(256), 0, stream>>>(W, M, out);
}